// WeightFn_1176821039589
// MI455X (gfx1250) — compile-verified
//
#include <hip/hip_runtime.h>
#include <math.h>

typedef __attribute__((ext_vector_type(2))) float v2f;
typedef __attribute__((ext_vector_type(4))) float v4f;
typedef __attribute__((ext_vector_type(8))) float v8f;

#define BB   32
#define MM   16384
#define WW   256
#define HH   16
#define EPSF 1e-5f
#define MB   256              // rows of M per workgroup
#define NBLK (MM / MB)        // 64 blocks along M

// -------- Kernel 1: WMMA logits + per-block softmax partials ---------------
__global__ __launch_bounds__(256) void k_logits(
    const float* __restrict__ mem, const float* __restrict__ keys,
    const float* __restrict__ strengths, float* __restrict__ out,
    float* __restrict__ pmax, float* __restrict__ psum)
{
    __shared__ float sk[HH][WW + 4];   // keys tile (padded)
    __shared__ float skn[HH];          // key norms
    __shared__ float ssp[HH];          // softplus(strength)
    __shared__ float tmax[16][HH];     // per-tile, per-h partial max
    __shared__ float tsum[16][HH];     // per-tile, per-h partial sum(exp)

    const int t     = threadIdx.x;
    const int b     = blockIdx.y;
    const int mbase = blockIdx.x * MB;
    const int lane  = t & 31;
    const int wv    = t >> 5;          // wave id 0..7
    const int half  = lane >> 4;       // 0 | 1
    const int l16   = lane & 15;

    // ---- stage keys[b] (16x256 f32) into LDS, coalesced ----
    const float* kb = keys + (size_t)b * HH * WW;
    #pragma unroll
    for (int j = 0; j < (HH * WW) / 256; ++j) {
        int idx = t + j * 256;
        sk[idx >> 8][idx & (WW - 1)] = kb[idx];
    }
    __syncthreads();

    if (t < HH) {
        float s = 0.f;
        for (int c = 0; c < WW; ++c) { float v = sk[t][c]; s += v * v; }
        skn[t] = sqrtf(s + EPSF);
        float st = strengths[b * HH + t];
        ssp[t] = (st > 20.f) ? st : log1pf(expf(st));   // softplus
    }
    __syncthreads();

    // ---- each wave: two 16x16 output tiles (h x m) ----
    for (int tt = 0; tt < 2; ++tt) {
        const int tile = wv * 2 + tt;
        const int m0   = mbase + tile * 16;
        // B-fragment source: memory row (m0 + l16), column offset 2*half
        const float* mrow = mem + ((size_t)b * MM + (size_t)(m0 + l16)) * WW + 2 * half;

        v8f acc = {0.f, 0.f, 0.f, 0.f, 0.f, 0.f, 0.f, 0.f};
        float sq = 0.f;

        #pragma unroll 8
        for (int k0 = 0; k0 < WW; k0 += 4) {
            v2f bf = *(const v2f*)(mrow + k0);          // memory^T 4x16 frag
            v2f af;                                      // keys 16x4 frag
            af.x = sk[l16][k0 + 2 * half];
            af.y = sk[l16][k0 + 2 * half + 1];
            sq += bf.x * bf.x + bf.y * bf.y;             // fused norm pass
            acc = __builtin_amdgcn_wmma_f32_16x16x4_f32(
                      false, af, false, bf, (short)0, acc, false, false);
        }

        // full row sumsq: combine the two 16-lane halves
        float rowsq = sq + __shfl_xor(sq, 16, 32);
        float mnorm = sqrtf(rowsq + EPSF);

        // scale -> logits; store; per-h (max, sumexp) partials
        float vals[8];
        #pragma unroll
        for (int i = 0; i < 8; ++i) {
            int h   = i + 8 * half;
            float v = acc[i] / (skn[h] * mnorm + EPSF) * ssp[h];
            vals[i] = v;
            out[((size_t)b * HH + h) * MM + (size_t)(m0 + l16)] = v;
        }
        #pragma unroll
        for (int i = 0; i < 8; ++i) {
            float mx = vals[i];
            #pragma unroll
            for (int d = 1; d < 16; d <<= 1) mx = fmaxf(mx, __shfl_xor(mx, d, 16));
            float e = expf(vals[i] - mx);
            float s = e;
            #pragma unroll
            for (int d = 1; d < 16; d <<= 1) s += __shfl_xor(s, d, 16);
            if (l16 == 0) {
                int h = i + 8 * half;
                tmax[tile][h] = mx;
                tsum[tile][h] = s;
            }
        }
    }
    __syncthreads();

    // ---- merge 16 tiles -> one per-block partial per h (fixed order) ----
    if (t < HH) {
        float gm = tmax[0][t];
        for (int k = 1; k < 16; ++k) gm = fmaxf(gm, tmax[k][t]);
        float gs = 0.f;
        for (int k = 0; k < 16; ++k) gs += tsum[k][t] * expf(tmax[k][t] - gm);
        int bh = b * HH + t;
        pmax[(size_t)bh * NBLK + blockIdx.x] = gm;
        psum[(size_t)bh * NBLK + blockIdx.x] = gs;
    }
}

// -------- Kernel 2: merge block partials per (b,h) -------------------------
__global__ void k_reduce(const float* __restrict__ pmax,
                         const float* __restrict__ psum,
                         float* __restrict__ gmax, float* __restrict__ gsum)
{
    int bh = blockIdx.x * blockDim.x + threadIdx.x;
    if (bh >= BB * HH) return;
    const float* pm = pmax + (size_t)bh * NBLK;
    const float* ps = psum + (size_t)bh * NBLK;
    float gm = pm[0];
    for (int k = 1; k < NBLK; ++k) gm = fmaxf(gm, pm[k]);
    float gs = 0.f;
    for (int k = 0; k < NBLK; ++k) gs += ps[k] * expf(pm[k] - gm);
    gmax[bh] = gm;
    gsum[bh] = 1.0f / gs;
}

// -------- Kernel 3: in-place softmax normalize, one block per (b,h) row ----
__global__ __launch_bounds__(256) void k_norm(float* __restrict__ out,
    const float* __restrict__ gmax, const float* __restrict__ gsum)
{
    int bh    = blockIdx.x;
    float gm  = gmax[bh];
    float inv = gsum[bh];
    v4f* row  = (v4f*)(out + (size_t)bh * MM);
    #pragma unroll
    for (int j = 0; j < (MM / 4) / 256; ++j) {
        int i = threadIdx.x + j * 256;
        v4f v = row[i];
        v.x = expf(v.x - gm) * inv;
        v.y = expf(v.y - gm) * inv;
        v.z = expf(v.z - gm) * inv;
        v.w = expf(v.w - gm) * inv;
        row[i] = v;
    }
}

extern "C" void kernel_launch(void* const* d_in, const int* in_sizes, int n_in,
                              void* d_out, int out_size, void* d_ws, size_t ws_size,
                              hipStream_t stream) {
    const float* mem       = (const float*)d_in[0];   // [B, M, W]
    const float* keys      = (const float*)d_in[1];   // [B, H, W]
    const float* strengths = (const float*)d_in[2];   // [B, H]
    float* out = (float*)d_out;                       // [B, H, M]

    float* ws   = (float*)d_ws;
    float* pmax = ws;                                 // [B*H][NBLK]
    float* psum = ws + (size_t)BB * HH * NBLK;        // [B*H][NBLK]
    float* gmax = ws + (size_t)2 * BB * HH * NBLK;    // [B*H]
    float* gsum = gmax + BB * HH;                     // [B*H]

    k_logits<<<dim3(NBLK, BB), 256, 0, stream>>>(mem, keys, strengths, out, pmax, psum);
    k_reduce<<<dim3(2), 256, 0, stream>>>(pmax, psum, gmax, gsum);
    k_norm  <<<dim3(BB * HH), 256, 0, stream>>>(out, gmax, gsum);
}